// Model_25769803776517
// MI455X (gfx1250) — compile-verified
//
#include <hip/hip_runtime.h>
#include <math.h>

#define N_FREQ   50
#define ENC_DIM  200
#define DEMB     5
#define N_GENES  1000

typedef float v2f __attribute__((ext_vector_type(2)));
typedef float v8f __attribute__((ext_vector_type(8)));

// ---------------------------------------------------------------------------
// Kernel 1: fused sine-encode + per-gene weight gather + contract + sigmoid.
// One thread per fragment: 100 sincos + 1000 FMA; weights (4MB) live in L2.
// ---------------------------------------------------------------------------
__global__ void frag_embed(const float* __restrict__ coords,
                           const int*   __restrict__ genemap,
                           const float* __restrict__ w1,   // [G,200,5]
                           float* __restrict__ emb,        // [F,5]
                           int nfrag)
{
    int f = blockIdx.x * blockDim.x + threadIdx.x;
    if (f >= nfrag) return;
    const float c01[2] = { coords[2 * f], coords[2 * f + 1] };
    const float* __restrict__ w = w1 + (size_t)genemap[f] * (ENC_DIM * DEMB);
    float acc[DEMB] = {0.f, 0.f, 0.f, 0.f, 0.f};
    const float r = 0.7585775750291836f;      // 1000^(-2/50)
    #pragma unroll
    for (int c = 0; c < 2; ++c) {
        float base = r;
        const float cc = c01[c];
        for (int k = 0; k < N_FREQ; ++k) {
            float s, co;
            __sincosf(cc * base, &s, &co);
            const float* wp = w + (c * 100 + 2 * k) * DEMB;
            #pragma unroll
            for (int d = 0; d < DEMB; ++d)
                acc[d] += s * wp[d] + co * wp[d + DEMB];
            base *= r;
        }
    }
    #pragma unroll
    for (int d = 0; d < DEMB; ++d)
        emb[(size_t)f * DEMB + d] = 1.f / (1.f + __expf(-acc[d]));
}

// ---------------------------------------------------------------------------
// Kernel 2: ragged self-attention via block-diagonal packing into 16x16 WMMA
// tiles.  One wave per tile (16 fragment rows = 16/gsz groups, block-diagonal
// mask kills cross-group terms).  scores: 2 chained V_WMMA_F32_16X16X4_F32
// (K = 8, D=5 zero-padded); apply: 4 chained WMMAs (K = 16 of W).
// ---------------------------------------------------------------------------
__global__ void attn_wmma(const int* __restrict__ idx, int ngroups, int gsz,
                          float* __restrict__ emb)
{
    constexpr int WPB = 4;                      // waves per block
    __shared__ float sx[WPB][16][8];            // X tile, dims 5..7 zero
    __shared__ float sw[WPB][16][16];           // softmax weights
    __shared__ int   srow[WPB][16];             // fragment id per row (-1 bad)
    __shared__ int   sgrp[WPB][16];             // group id per row (neg bad)

    const int wave = threadIdx.x >> 5;
    const int lane = threadIdx.x & 31;
    const int half = lane >> 4;
    const int l16  = lane & 15;

    const int gpt    = 16 / gsz;                // groups per tile
    const int rows   = gpt * gsz;               // valid rows per tile
    const int tile   = blockIdx.x * WPB + wave;
    const int ntiles = (ngroups + gpt - 1) / gpt;
    if (tile >= ntiles) return;                 // wave-uniform exit

    // ---- stage X tile into LDS ------------------------------------------
    for (int i = lane; i < 16 * 8; i += 32)
        (&sx[wave][0][0])[i] = 0.f;
    if (lane < 16) {
        int rrow  = lane;
        int gi    = tile * gpt + rrow / gsz;
        int valid = (rrow < rows) && (gi < ngroups);
        int f     = valid ? idx[gi * gsz + rrow % gsz] : -1;
        srow[wave][rrow] = f;
        sgrp[wave][rrow] = valid ? gi : (-1 - rrow); // unique tag if invalid
        if (valid) {
            #pragma unroll
            for (int d = 0; d < DEMB; ++d)
                sx[wave][rrow][d] = emb[(size_t)f * DEMB + d];
        }
    }
    asm volatile("s_wait_dscnt 0" ::: "memory");

    // ---- scores = X * X^T  (A and B register images coincide) -----------
    v2f a0, a1;
    a0.x = sx[wave][l16][2 * half + 0];
    a0.y = sx[wave][l16][2 * half + 1];
    a1.x = sx[wave][l16][4 + 2 * half + 0];
    a1.y = sx[wave][l16][4 + 2 * half + 1];
    v8f c = {0.f, 0.f, 0.f, 0.f, 0.f, 0.f, 0.f, 0.f};
    c = __builtin_amdgcn_wmma_f32_16x16x4_f32(false, a0, false, a0,
                                              (short)0, c, false, false);
    c = __builtin_amdgcn_wmma_f32_16x16x4_f32(false, a1, false, a1,
                                              (short)0, c, false, false);

    // ---- masked softmax per row (reduce across the 16-lane half) --------
    const float inv_sqrt = rsqrtf((float)gsz);  // reference scales by sqrt(seq)
    const int cg = sgrp[wave][l16];             // column group
    float wv[8];
    #pragma unroll
    for (int v = 0; v < 8; ++v) {
        int   M  = v + 8 * half;
        int   rg = sgrp[wave][M];
        bool  ok = (rg >= 0) && (rg == cg);
        float sc = ok ? c[v] * inv_sqrt : -INFINITY;
        float mx = sc;
        #pragma unroll
        for (int m = 8; m >= 1; m >>= 1)
            mx = fmaxf(mx, __shfl_xor(mx, m, 32));
        float e = ok ? __expf(sc - mx) : 0.f;
        float sum = e;
        #pragma unroll
        for (int m = 8; m >= 1; m >>= 1)
            sum += __shfl_xor(sum, m, 32);
        wv[v] = (sum > 0.f) ? (e / sum) : 0.f;
    }

    // ---- stage W through LDS to rebuild it in A-operand layout ----------
    #pragma unroll
    for (int v = 0; v < 8; ++v)
        sw[wave][v + 8 * half][l16] = wv[v];
    asm volatile("s_wait_dscnt 0" ::: "memory");

    // ---- out = W (16x16) * X (16x5)  : 4 chained WMMAs over K=16 --------
    v8f o = {0.f, 0.f, 0.f, 0.f, 0.f, 0.f, 0.f, 0.f};
    #pragma unroll
    for (int k = 0; k < 4; ++k) {
        v2f a, b;
        a.x = sw[wave][l16][4 * k + 2 * half + 0];
        a.y = sw[wave][l16][4 * k + 2 * half + 1];
        b.x = (l16 < 8) ? sx[wave][4 * k + 2 * half + 0][l16] : 0.f;
        b.y = (l16 < 8) ? sx[wave][4 * k + 2 * half + 1][l16] : 0.f;
        o = __builtin_amdgcn_wmma_f32_16x16x4_f32(false, a, false, b,
                                                  (short)0, o, false, false);
    }

    // ---- scatter results back (C layout: lane l16 = dim, vgpr v = row) --
    if (l16 < DEMB) {
        #pragma unroll
        for (int v = 0; v < 8; ++v) {
            int M = v + 8 * half;
            int f = srow[wave][M];
            if (f >= 0) emb[(size_t)f * DEMB + l16] = o[v];
        }
    }
}

// ---------------------------------------------------------------------------
// Kernel 3: initialize output with the per-gene bias broadcast.
// ---------------------------------------------------------------------------
__global__ void out_init(const int* __restrict__ genes_oi,
                         const float* __restrict__ bias1,
                         float* __restrict__ out, int total)
{
    int s = blockIdx.x * blockDim.x + threadIdx.x;
    if (s >= total) return;
    int g = s % N_GENES;
    out[s] = bias1[genes_oi[g]];
}

// ---------------------------------------------------------------------------
// Kernel 4: fused segment-sum + expression head.  Each fragment adds a
// scalar dot(emb_f, w_expr[gene]) into out[cell*G+gene] with fp32 HW atomics.
// Avoids ever materializing the 100 MB pooled tensor.
// ---------------------------------------------------------------------------
__global__ void pool_scatter(const float* __restrict__ emb,
                             const int*   __restrict__ lcx,
                             const int*   __restrict__ genes_oi,
                             const float* __restrict__ wexpr,  // [G,5]
                             float* __restrict__ out, int nfrag)
{
    int f = blockIdx.x * blockDim.x + threadIdx.x;
    if (f >= nfrag) return;
    int s = lcx[f];
    int g = s % N_GENES;
    const float* we = wexpr + (size_t)genes_oi[g] * DEMB;
    float dot = 0.f;
    #pragma unroll
    for (int d = 0; d < DEMB; ++d)
        dot += emb[(size_t)f * DEMB + d] * we[d];
    __hip_atomic_fetch_add(&out[s], dot, __ATOMIC_RELAXED,
                           __HIP_MEMORY_SCOPE_AGENT);
}

// ---------------------------------------------------------------------------
extern "C" void kernel_launch(void* const* d_in, const int* in_sizes, int n_in,
                              void* d_out, int out_size, void* d_ws, size_t ws_size,
                              hipStream_t stream)
{
    const float* coords   = (const float*)d_in[0];
    const int*   genemap  = (const int*)  d_in[1];
    const int*   n2       = (const int*)  d_in[2];
    const int*   n3       = (const int*)  d_in[3];
    const int*   n4       = (const int*)  d_in[4];
    const int*   lcx      = (const int*)  d_in[5];
    const int*   genes_oi = (const int*)  d_in[6];
    const float* w1       = (const float*)d_in[7];
    const float* wexpr    = (const float*)d_in[8];
    const float* bias1    = (const float*)d_in[9];
    float*       out      = (float*)d_out;
    float*       emb      = (float*)d_ws;          // [nfrag, 5] scratch

    const int nfrag = in_sizes[0] / 2;

    frag_embed<<<(nfrag + 255) / 256, 256, 0, stream>>>(coords, genemap, w1,
                                                        emb, nfrag);

    const int*  segidx[3] = { n2, n3, n4 };
    const int   segcnt[3] = { in_sizes[2], in_sizes[3], in_sizes[4] };
    const int   seggsz[3] = { 2, 3, 4 };
    for (int i = 0; i < 3; ++i) {
        int gsz     = seggsz[i];
        int ngroups = segcnt[i] / gsz;
        int gpt     = 16 / gsz;
        int ntiles  = (ngroups + gpt - 1) / gpt;
        int blocks  = (ntiles + 3) / 4;            // 4 waves (tiles) per block
        attn_wmma<<<blocks, 128, 0, stream>>>(segidx[i], ngroups, gsz, emb);
    }

    out_init<<<(out_size + 255) / 256, 256, 0, stream>>>(genes_oi, bias1,
                                                         out, out_size);
    pool_scatter<<<(nfrag + 255) / 256, 256, 0, stream>>>(emb, lcx, genes_oi,
                                                          wexpr, out, nfrag);
}